// InplaceTopK_23321672417516
// MI455X (gfx1250) — compile-verified
//
#include <hip/hip_runtime.h>
#include <stdint.h>

typedef uint32_t u32;
typedef uint64_t u64;

#define ROWS_PER_BLOCK 192      // 6 wave32s; rows per workgroup (thread-per-row)
#define NCAND 128
#define TOPK 16
#define CHUNK 32                // candidates per TDM chunk
#define NCHUNK (NCAND / CHUNK)  // 4
#define LDS_PITCH 33            // 32 payload dwords + 1 pad dword (TDM pad) -> conflict-free

// ---- CDNA5 TDM availability probe (device pass only) ----
#if defined(__AMDGCN__) && defined(__gfx1250__) && defined(__has_builtin)
# if __has_builtin(__builtin_amdgcn_tensor_load_to_lds) && __has_builtin(__builtin_amdgcn_s_wait_tensorcnt)
#  define HAVE_TDM 1
# endif
#endif

typedef unsigned int v4u __attribute__((ext_vector_type(4)));
typedef int          v8i __attribute__((ext_vector_type(8)));
typedef int          v4i __attribute__((ext_vector_type(4)));

// monotonic float->u32 key (order-preserving for all non-NaN floats)
__device__ __forceinline__ u32 f2key(float f) {
    u32 u = __float_as_uint(f);
    return u ^ ((u32)((int)u >> 31) | 0x80000000u);
}
__device__ __forceinline__ float key2f(u32 s) {
    u32 u = s ^ (((int)s < 0) ? 0x80000000u : 0xFFFFFFFFu);
    return __uint_as_float(u);
}

// sorted-descending insertion into 16-entry register list of 64-bit keys.
// Chain itself is fully unrolled (constant indices -> registers, no scratch);
// callers keep their loops ROLLED so this body appears only 5 times in the code (1 seed + 4 chunks).
__device__ __forceinline__ void insert16(u64 (&k)[TOPK], u64 ck) {
    if (ck > k[TOPK - 1]) {
        u64 cur = ck;
#pragma unroll
        for (int j = 0; j < TOPK; ++j) {
            bool sw  = cur > k[j];
            u64  hi  = sw ? cur  : k[j];   // stays at slot j
            u64  lo  = sw ? k[j] : cur;    // continues down
            k[j] = hi;
            cur  = lo;
        }
    }
}

#if defined(HAVE_TDM)
// Issue one TDM 2-D tile load: CHUNK f32 per row x ROWS_PER_BLOCK rows,
// row pitch NCAND elements in memory, padded to LDS_PITCH dwords in LDS.
__device__ __forceinline__ void tdm_load_chunk(const float* gbase, u32 lds_byte_addr,
                                               u32 rows_remaining) {
    v4u g0;
    v8i g1;
    v4i gz = {0, 0, 0, 0};
    u64 ga = (u64)(uintptr_t)gbase;
    // group0: count=1 | lds_addr | global_addr(57b) | type=2
    g0[0] = 1u;
    g0[1] = lds_byte_addr;
    g0[2] = (u32)ga;
    g0[3] = (u32)(ga >> 32) | (2u << 30);
    // group1: data_size=4B(2), pad_enable, pad_interval=32dw(4), pad_amount=1dw(0)
    g1[0] = (int)((2u << 16) | (1u << 20) | (4u << 22));
    g1[1] = (int)((u32)NCAND << 16);                            // tensor_dim0[15:0]
    g1[2] = (int)((rows_remaining & 0xFFFFu) << 16);            // dim0 hi=0 | dim1 lo
    g1[3] = (int)((rows_remaining >> 16) | ((u32)CHUNK << 16)); // dim1 hi | tile_dim0
    g1[4] = (int)ROWS_PER_BLOCK;                                // tile_dim1 | tile_dim2=0
    g1[5] = (int)NCAND;                                         // tensor_dim0_stride lo32
    g1[6] = 0;
    g1[7] = 0;
#if __clang_major__ >= 23
    v8i gz8 = {0, 0, 0, 0, 0, 0, 0, 0};
    __builtin_amdgcn_tensor_load_to_lds(g0, g1, gz, gz, gz8, 0);
#else
    __builtin_amdgcn_tensor_load_to_lds(g0, g1, gz, gz, 0);
#endif
}
#endif

__global__ __launch_bounds__(ROWS_PER_BLOCK)
void InplaceTopK_kernel(const float* __restrict__ cand,      // [rows,128]
                        const long long* __restrict__ cidx,  // [128]
                        const float* __restrict__ topv,      // [rows,16]
                        const long long* __restrict__ topi,  // [rows,16]
                        const float* __restrict__ sums,      // [rows]
                        const float* __restrict__ sqs,       // [rows]
                        const int* __restrict__ cnt,         // [1]
                        float* __restrict__ outv,
                        long long* __restrict__ outi,
                        int* __restrict__ outmin,
                        float* __restrict__ outsum,
                        float* __restrict__ outsq,
                        int* __restrict__ outcnt,
                        int rows) {
    __shared__ u32 lbuf[2][ROWS_PER_BLOCK * LDS_PITCH];

    const int  t    = threadIdx.x;
    const int  row0 = blockIdx.x * ROWS_PER_BLOCK;
    const int  r    = row0 + t;
    const bool active = (r < rows);

#if defined(HAVE_TDM)
    if (t < 32) {  // wave 0 drives the DMA engine (TDM ignores EXEC)
        const u32 rem = (u32)(rows - row0);
        tdm_load_chunk(cand + (size_t)row0 * NCAND + 0 * CHUNK,
                       (u32)(uintptr_t)&lbuf[0][0], rem);
        tdm_load_chunk(cand + (size_t)row0 * NCAND + 1 * CHUNK,
                       (u32)(uintptr_t)&lbuf[1][0], rem);
    }
#endif

    // seed the list with the existing top-k (concat positions 0..15).
    // Rolled loop: one insert16 site, ~16 dynamic iterations.
    u64 k[TOPK];
#pragma unroll
    for (int j = 0; j < TOPK; ++j) k[j] = 0;  // sentinel: below any real key

    if (active) {
        const float* tv = topv + (size_t)r * TOPK;
#pragma unroll 1
        for (int pos = 0; pos < TOPK; ++pos) {
            insert16(k, ((u64)f2key(tv[pos]) << 8) | (u32)(255 - pos));
        }
    }

    float acc = 0.0f, acc2 = 0.0f;

#pragma unroll
    for (int c = 0; c < NCHUNK; ++c) {
#if defined(HAVE_TDM)
        if (t < 32) {
            // builtin requires a literal immediate -> two call sites
            if (c < NCHUNK - 1) {
                __builtin_amdgcn_s_wait_tensorcnt(1);
            } else {
                __builtin_amdgcn_s_wait_tensorcnt(0);
            }
        }
#else
        // fallback: coalesced cooperative staging of chunk c into lbuf[c&1]
        for (int e = t; e < ROWS_PER_BLOCK * CHUNK; e += ROWS_PER_BLOCK) {
            const int rr = e >> 5, cc = e & 31;
            float x = ((row0 + rr) < rows)
                          ? cand[(size_t)(row0 + rr) * NCAND + c * CHUNK + cc]
                          : 0.0f;
            lbuf[c & 1][rr * LDS_PITCH + cc] = __float_as_uint(x);
        }
#endif
        __syncthreads();  // chunk c resident in lbuf[c&1]

        if (active) {
            const u32* lr = &lbuf[c & 1][t * LDS_PITCH];  // pitch 33 -> no bank conflicts
            // Rolled loop: one insert16 site per chunk, identical dynamic work,
            // ~10x smaller i-cache footprint than full unroll.
#pragma unroll 1
            for (int i = 0; i < CHUNK; ++i) {
                const float x = __uint_as_float(lr[i]);
                acc += x;
                acc2 = fmaf(x, x, acc2);
                const int pos = TOPK + c * CHUNK + i;
                insert16(k, ((u64)f2key(x) << 8) | (u32)(255 - pos));
            }
        }
        __syncthreads();  // everyone done reading lbuf[c&1]

#if defined(HAVE_TDM)
        if ((c + 2 < NCHUNK) && (t < 32)) {  // refill the just-freed buffer
            tdm_load_chunk(cand + (size_t)row0 * NCAND + (c + 2) * CHUNK,
                           (u32)(uintptr_t)&lbuf[c & 1][0], (u32)(rows - row0));
        }
#endif
    }

    if (active) {
        // Epilogue stays unrolled: register arrays need constant indices.
        float vals[TOPK];
#pragma unroll
        for (int j = 0; j < TOPK; ++j) vals[j] = key2f((u32)(k[j] >> 8));

        // sorted top values: 64B contiguous per thread
        float4* ov = (float4*)(outv + (size_t)r * TOPK);
        ov[0] = make_float4(vals[0], vals[1], vals[2], vals[3]);
        ov[1] = make_float4(vals[4], vals[5], vals[6], vals[7]);
        ov[2] = make_float4(vals[8], vals[9], vals[10], vals[11]);
        ov[3] = make_float4(vals[12], vals[13], vals[14], vals[15]);

        // recover int64 indices from the position byte (gathers hit hot lines)
        const long long* tir = topi + (size_t)r * TOPK;
        long long* oip = outi + (size_t)r * TOPK;
#pragma unroll
        for (int j = 0; j < TOPK; ++j) {
            const int pos = 255 - (int)(k[j] & 0xFFu);
            oip[j] = (pos < TOPK) ? tir[pos] : cidx[pos - TOPK];
        }

        // argmin = first occurrence of the minimum in the descending list
        int am = TOPK - 1;
#pragma unroll
        for (int j = TOPK - 2; j >= 0; --j) am = (vals[j] == vals[TOPK - 1]) ? j : am;
        outmin[r] = am;

        outsum[r] = sums[r] + acc;
        outsq[r]  = sqs[r] + acc2;
        if (r == 0) outcnt[0] = cnt[0] + NCAND;
    }
}

extern "C" void kernel_launch(void* const* d_in, const int* in_sizes, int n_in,
                              void* d_out, int out_size, void* d_ws, size_t ws_size,
                              hipStream_t stream) {
    (void)n_in; (void)out_size; (void)d_ws; (void)ws_size;
    const float*     cand = (const float*)d_in[0];
    const long long* cidx = (const long long*)d_in[1];
    const float*     topv = (const float*)d_in[2];
    const long long* topi = (const long long*)d_in[3];
    const float*     sums = (const float*)d_in[4];
    const float*     sqs  = (const float*)d_in[5];
    const int*       cnt  = (const int*)d_in[6];

    const int rows = in_sizes[4];  // len(sums) == SIZE

    // output layout (bytes), tuple members flattened in return order
    char*  ob   = (char*)d_out;
    size_t oTV  = 0;
    size_t oTI  = oTV + (size_t)rows * TOPK * sizeof(float);
    size_t oMIN = oTI + (size_t)rows * TOPK * sizeof(long long);
    size_t oSUM = oMIN + (size_t)rows * sizeof(int);
    size_t oSQ  = oSUM + (size_t)rows * sizeof(float);
    size_t oCNT = oSQ + (size_t)rows * sizeof(float);

    dim3 grid((rows + ROWS_PER_BLOCK - 1) / ROWS_PER_BLOCK);
    dim3 block(ROWS_PER_BLOCK);
    hipLaunchKernelGGL(InplaceTopK_kernel, grid, block, 0, stream,
                       cand, cidx, topv, topi, sums, sqs, cnt,
                       (float*)(ob + oTV), (long long*)(ob + oTI),
                       (int*)(ob + oMIN), (float*)(ob + oSUM),
                       (float*)(ob + oSQ), (int*)(ob + oCNT), rows);
}